// MCL_Loss_47253230191228
// MI455X (gfx1250) — compile-verified
//
#include <hip/hip_runtime.h>
#include <math.h>

typedef __attribute__((ext_vector_type(2))) float v2f;
typedef __attribute__((ext_vector_type(8))) float v8f;

#define NBR      4
#define REP_DIM  1024
#define FEAT_DIM 128
#define N_DATA   100000
#define NCE_K    4096
#define KTOT     4097          // NCE_K + 1
#define BATCH    128
#define INV_T    (1.0f / 0.07f)
#define M_PN     0.04096f                 // NCE_K / N_DATA
#define C_EPS    (0.04096f + 1e-7f)       // m*Pn + EPS
#define NEG_CNT  524288.0f                // BATCH * NCE_K
#define TOT_CNT  524416.0f                // BATCH * KTOT

// workspace layout (floats)
#define WS_E      0                       // 4*128*128 = 65536 floats (e vectors)
#define WS_SUMEXP (4 * 128 * 128)         // 16 floats
#define WS_G      (4 * 128 * 128 + 16)    // 16 floats
#define WS_P      (4 * 128 * 128 + 32)    // 16 floats

// ---------------------------------------------------------------------------
// Stage 0: zero the 48 accumulator floats (SumExp, G, P are contiguous).
// ---------------------------------------------------------------------------
__global__ void stage0_init(float* __restrict__ p) {
  if (threadIdx.x < 48) p[threadIdx.x] = 0.0f;
}

// ---------------------------------------------------------------------------
// Stage 1: h[r,b,d] = sum_k emb_r[b,k] * W[r,d,k] + bias[r,d]
// f32 WMMA 16x16x4, one wave per 16x16 output tile. 4 branches * 8*8 tiles
// = 256 waves = 32 blocks of 256 threads.
// Branch pointer selected via ternary chain (not an indexed local array) so
// infer-address-spaces keeps it in addrspace(1) -> global_load_b64, not flat.
// ---------------------------------------------------------------------------
__global__ __launch_bounds__(256) void stage1_gemm(
    const float* __restrict__ e0, const float* __restrict__ e1,
    const float* __restrict__ e2, const float* __restrict__ e3,
    const float* __restrict__ W, const float* __restrict__ bias,
    float* __restrict__ eBuf) {
  int wave = (blockIdx.x * blockDim.x + threadIdx.x) >> 5;
  int lane = threadIdx.x & 31;
  int r  = wave >> 6;
  int mT = (wave >> 3) & 7;
  int nT = wave & 7;
  int li = lane & 15, half = lane >> 4;

  const float* emb = (r == 0) ? e0 : (r == 1) ? e1 : (r == 2) ? e2 : e3;

  // A[M=li, K = k + 2*half + v] = emb[mT*16+li, k...]
  const float* Ap = emb + (size_t)(mT * 16 + li) * REP_DIM + 2 * half;
  // B[K, N=li] = W[r, nT*16+li, K]
  const float* Bp = W + (size_t)r * FEAT_DIM * REP_DIM +
                    (size_t)(nT * 16 + li) * REP_DIM + 2 * half;
  v8f c = {};
  for (int k = 0; k < REP_DIM; k += 4) {
    v2f a = *(const v2f*)(Ap + k);
    v2f b = *(const v2f*)(Bp + k);
    c = __builtin_amdgcn_wmma_f32_16x16x4_f32(false, a, false, b, (short)0, c,
                                              false, false);
  }
  int n = nT * 16 + li;
  float bv = bias[r * FEAT_DIM + n];
#pragma unroll
  for (int v = 0; v < 8; ++v) {
    int m = mT * 16 + v + 8 * half;                   // D row
    eBuf[(size_t)(r * BATCH + m) * FEAT_DIM + n] = c[v] + bv;
  }
}

// ---------------------------------------------------------------------------
// Stage 2: L2-normalize each 128-float row of eBuf in place. One wave/row.
// ---------------------------------------------------------------------------
__global__ __launch_bounds__(256) void stage2_norm(float* __restrict__ eBuf) {
  int wave = (blockIdx.x * blockDim.x + threadIdx.x) >> 5;
  int lane = threadIdx.x & 31;
  float* row = eBuf + (size_t)wave * FEAT_DIM;
  float4 x = *(float4*)(row + lane * 4);
  float s = x.x * x.x + x.y * x.y + x.z * x.z + x.w * x.w;
  for (int m = 16; m > 0; m >>= 1) s += __shfl_xor(s, m, 32);
  float inv = 1.0f / sqrtf(s);
  x.x *= inv; x.y *= inv; x.z *= inv; x.w *= inv;
  *(float4*)(row + lane * 4) = x;
}

// ---------------------------------------------------------------------------
// Stage 3: scores s[i,j,b,k] = dot(memory[j, cidx[b,k], :], e[i,b,:]).
// One wave handles one (b, 4-wide k-chunk) tile:
//   A rows (M=16) = 4 j's x 4 k's of gathered memory rows, reduced over d=128
//   B cols (N)    = the 4 e-vectors e[i,b,:] (cols 4..15 replicate, ignored)
// D layout puts each (i,j) partial in exactly one lane -> LDS f32 atomics ->
// one global f32 atomic per block per (i,j).
// PASS 0: accumulate SumExp[i,j] = sum exp(s/T)
// PASS 1: accumulate G[i,j] = sum log(exp(s/T)/Z + c), P[i,j] = sum_b s0/T
// Total waves: 128 * 1025 = 131200 -> 16400 blocks of 256 threads.
// ---------------------------------------------------------------------------
template <int PASS>
__global__ __launch_bounds__(256) void stage3_scores(
    const float* __restrict__ memory, const int* __restrict__ cidx,
    const float* __restrict__ eBuf, float* __restrict__ sumExp,
    float* __restrict__ gOut, float* __restrict__ pOut) {
  __shared__ float accA[16];
  __shared__ float accB[16];
  int tid = threadIdx.x;
  if (tid < 16) { accA[tid] = 0.0f; accB[tid] = 0.0f; }
  __syncthreads();

  int wave = (blockIdx.x * blockDim.x + tid) >> 5;
  int lane = tid & 31;
  int b  = wave / 1025;
  int kc = wave % 1025;
  int kBase = kc * 4;
  int li = lane & 15, half = lane >> 4;

  // A-fragment: row m = li -> (j = li>>2, k' = kBase + (li&3)); clamp OOB k'.
  int j  = li >> 2;
  int kp = kBase + (li & 3);
  if (kp > NCE_K) kp = NCE_K;
  int rowIdx = cidx[b * KTOT + kp];
  const float* Ap =
      memory + ((size_t)j * N_DATA + (size_t)rowIdx) * FEAT_DIM + 2 * half;
  // B-fragment: col n = li -> e[i = li&3, b, :] (cols >=4 replicate; ignored).
  const float* Bp =
      eBuf + (size_t)((li & 3) * BATCH + b) * FEAT_DIM + 2 * half;

  v8f c = {};
  for (int d = 0; d < FEAT_DIM; d += 4) {
    v2f a  = *(const v2f*)(Ap + d);
    v2f bb = *(const v2f*)(Bp + d);
    c = __builtin_amdgcn_wmma_f32_16x16x4_f32(false, a, false, bb, (short)0, c,
                                              false, false);
  }

  // Lane (li<4): i = li; c[v] = s[i, j = (v+8*half)>>2, b, k = kBase+(v&3)]
  if (li < 4) {
    int i = li;
    int j0 = 2 * half;                       // v = 0..3
    int j1 = 2 * half + 1;                   // v = 4..7
    if (PASS == 0) {
      float s0 = 0.0f, s1 = 0.0f;
#pragma unroll
      for (int v = 0; v < 8; ++v) {
        int kk = kBase + (v & 3);
        float ev = (kk < KTOT) ? expf(c[v] * INV_T) : 0.0f;
        if (v < 4) s0 += ev; else s1 += ev;
      }
      atomicAdd(&accA[i * 4 + j0], s0);
      atomicAdd(&accA[i * 4 + j1], s1);
    } else {
      float iZ0 = TOT_CNT / ((float)N_DATA * sumExp[i * 4 + j0]);
      float iZ1 = TOT_CNT / ((float)N_DATA * sumExp[i * 4 + j1]);
      float g0 = 0.0f, g1 = 0.0f, p0 = 0.0f, p1 = 0.0f;
#pragma unroll
      for (int v = 0; v < 8; ++v) {
        int kk = kBase + (v & 3);
        float sc = c[v] * INV_T;
        float t  = expf(sc) * ((v < 4) ? iZ0 : iZ1);
        float g  = (kk < KTOT) ? logf(t + C_EPS) : 0.0f;
        if (v < 4) g0 += g; else g1 += g;
        if (kk == 0) { if (v < 4) p0 += sc; else p1 += sc; }  // positive term
      }
      atomicAdd(&accA[i * 4 + j0], g0);
      atomicAdd(&accA[i * 4 + j1], g1);
      atomicAdd(&accB[i * 4 + j0], p0);
      atomicAdd(&accB[i * 4 + j1], p1);
    }
  }
  __syncthreads();
  if (tid < 16) {
    if (PASS == 0) {
      atomicAdd(&sumExp[tid], accA[tid]);
    } else {
      atomicAdd(&gOut[tid], accA[tid]);
      atomicAdd(&pOut[tid], accB[tid]);
    }
  }
}

// ---------------------------------------------------------------------------
// Stage 4: loss = sum_{i!=j} -(P - B*logZ + B*K*log(mPn) - G) / B
// ---------------------------------------------------------------------------
__global__ void stage4_final(const float* __restrict__ sumExp,
                             const float* __restrict__ gAcc,
                             const float* __restrict__ pAcc,
                             float* __restrict__ out) {
  int t = threadIdx.x;
  float val = 0.0f;
  if (t < 16) {
    int i = t >> 2, j = t & 3;
    float Z    = sumExp[t] * ((float)N_DATA / TOT_CNT);
    float tot  = pAcc[t] - (float)BATCH * logf(Z) + NEG_CNT * logf(M_PN) -
                 gAcc[t];
    float per  = -tot / (float)BATCH;
    val = (i != j) ? per : 0.0f;
  }
  for (int m = 16; m > 0; m >>= 1) val += __shfl_xor(val, m, 32);
  if (t == 0) out[0] = val;
}

// ---------------------------------------------------------------------------
extern "C" void kernel_launch(void* const* d_in, const int* in_sizes, int n_in,
                              void* d_out, int out_size, void* d_ws,
                              size_t ws_size, hipStream_t stream) {
  (void)in_sizes; (void)n_in; (void)out_size; (void)ws_size;
  const float* e0     = (const float*)d_in[0];
  const float* e1     = (const float*)d_in[1];
  const float* e2     = (const float*)d_in[2];
  const float* e3     = (const float*)d_in[3];
  const float* W      = (const float*)d_in[4];
  const float* bias   = (const float*)d_in[5];
  const float* memory = (const float*)d_in[6];
  // d_in[7] = idx (redundant: equals contrast_idx[:,0])
  const int* cidx     = (const int*)d_in[8];

  float* ws     = (float*)d_ws;
  float* eBuf   = ws + WS_E;
  float* sumExp = ws + WS_SUMEXP;
  float* gAcc   = ws + WS_G;
  float* pAcc   = ws + WS_P;

  stage0_init<<<1, 64, 0, stream>>>(sumExp);
  stage1_gemm<<<32, 256, 0, stream>>>(e0, e1, e2, e3, W, bias, eBuf);
  stage2_norm<<<64, 256, 0, stream>>>(eBuf);              // 512 rows
  stage3_scores<0><<<16400, 256, 0, stream>>>(memory, cidx, eBuf, sumExp,
                                              gAcc, pAcc);
  stage3_scores<1><<<16400, 256, 0, stream>>>(memory, cidx, eBuf, sumExp,
                                              gAcc, pAcc);
  stage4_final<<<1, 32, 0, stream>>>(sumExp, gAcc, pAcc, (float*)d_out);
}